// DFTQNN_59760174957168
// MI455X (gfx1250) — compile-verified
//
#include <hip/hip_runtime.h>

typedef __attribute__((ext_vector_type(2))) float v2f;
typedef __attribute__((ext_vector_type(8))) float v8f;

#define DIM     1024
#define NGATES  8
#define NTERMS  40   // Taylor terms; |t*lambda| <~ 5 -> 40 terms is far past fp32 convergence

// ---------------------------------------------------------------------------
// init: psi = feature / ||feature||  (complex: imag = 0), w = psi
// ---------------------------------------------------------------------------
__global__ void __launch_bounds__(256) qnn_init(const float* __restrict__ feature,
                                                float* __restrict__ psi_r,
                                                float* __restrict__ psi_i,
                                                float* __restrict__ w_r,
                                                float* __restrict__ w_i) {
    __shared__ float red[256];
    const int tid = threadIdx.x;
    float s = 0.f;
    for (int i = tid; i < DIM; i += 256) { float v = feature[i]; s += v * v; }
    red[tid] = s;
    __syncthreads();
    for (int off = 128; off > 0; off >>= 1) {
        if (tid < off) red[tid] += red[tid + off];
        __syncthreads();
    }
    const float inv = rsqrtf(red[0]);
    for (int i = tid; i < DIM; i += 256) {
        float v = feature[i] * inv;
        psi_r[i] = v; psi_i[i] = 0.f;
        w_r[i]   = v; w_i[i]   = 0.f;
    }
}

// ---------------------------------------------------------------------------
// gate begin: w0 = psi
// ---------------------------------------------------------------------------
__global__ void __launch_bounds__(256) qnn_copy(const float* __restrict__ psi_r,
                                                const float* __restrict__ psi_i,
                                                float* __restrict__ w_r,
                                                float* __restrict__ w_i) {
    const int i = blockIdx.x * 256 + threadIdx.x;
    w_r[i] = psi_r[i];
    w_i[i] = psi_i[i];
}

// ---------------------------------------------------------------------------
// One Taylor term:  y = G @ w_in   (real 1024x1024 times [w_r w_i])
//                   w_out = (t/k) * ( y_i, -y_r )     [ = (-i t/k) * y ]
//                   psi  += w_out
// Block = 256 threads (8 waves), owns 16 output rows; waves split K 8 ways.
// WMMA f32 16x16x4: A = G tile (16 rows x 4 k), B = w tile (4 k x 16 cols).
// Only B columns 0(re)/1(im) carry data; lanes for columns >= 2 read from a
// zeroed LDS row so the B load is a plain unconditional ds_load_b64 (no EXEC
// predication in the hot loop).
// ---------------------------------------------------------------------------
__global__ void __launch_bounds__(256)
qnn_term(const float* __restrict__ G,
         const float* __restrict__ theta_g,
         const float* __restrict__ win_r, const float* __restrict__ win_i,
         float* __restrict__ wout_r, float* __restrict__ wout_i,
         float* __restrict__ psi_r,  float* __restrict__ psi_i,
         float inv_k) {
    __shared__ float lw[3 * DIM];       // [0]=w_r, [1]=w_i, [2]=zeros : 12 KB
    __shared__ float part[8][2][16];    // per-wave partial y tiles (n=re/im, m=row)
    __shared__ float ysum[2][16];

    const int tid     = threadIdx.x;
    const int wave    = tid >> 5;
    const int lane    = tid & 31;
    const int rowBase = blockIdx.x * 16;

    // stage w into LDS (read by every wave) + zero row for idle B columns
    for (int i = tid; i < DIM; i += 256) {
        lw[0 * DIM + i] = win_r[i];
        lw[1 * DIM + i] = win_i[i];
        lw[2 * DIM + i] = 0.f;
    }
    __syncthreads();

    const int m    = lane & 15;     // A: row in tile; B: column in tile
    const int half = lane >> 4;     // selects K pair {0,1} vs {2,3}
    const int koff = half * 2;

    const float* __restrict__ Grow = G + (size_t)(rowBase + m) * DIM;
    const int brow = (m < 2) ? m : 2;                 // zero row for cols >= 2
    const float* __restrict__ Bsrc = &lw[brow * DIM];

    v8f acc = {};
    const int kstart = wave * (DIM / 8);    // 128-wide K chunk per wave

#pragma unroll 8
    for (int kk = 0; kk < DIM / 8; kk += 4) {
        const int kb = kstart + kk;
        // A operand: lane holds G[rowBase+m][kb+koff], G[rowBase+m][kb+koff+1]
        v2f a;
        a.x = Grow[kb + koff];
        a.y = Grow[kb + koff + 1];
        // B operand: lane holds w[kb+koff][n], w[kb+koff+1][n] (zeros if n>=2)
        v2f b;
        b.x = Bsrc[kb + koff];
        b.y = Bsrc[kb + koff + 1];
        // v_wmma_f32_16x16x4_f32 : D = A*B + C  (EXEC uniform here)
        acc = __builtin_amdgcn_wmma_f32_16x16x4_f32(
            false, a, false, b, (short)0, acc, false, false);
    }

    // D layout: vgpr j, lane l -> row = 8*(l/16)+j, col = l%16 (only col 0/1 used)
    if (m < 2) {
#pragma unroll
        for (int j = 0; j < 8; ++j)
            part[wave][m][half * 8 + j] = acc[j];
    }
    __syncthreads();

    // reduce the 8 K-chunk partials
    if (tid < 32) {
        const int n = tid >> 4, mm = tid & 15;
        float s = 0.f;
#pragma unroll
        for (int w = 0; w < 8; ++w) s += part[w][n][mm];
        ysum[n][mm] = s;
    }
    __syncthreads();

    // coefficient rotation + psi accumulation (fused epilogue)
    if (tid < 16) {
        const float s  = theta_g[0] * inv_k;     // t / k
        const float yr = ysum[0][tid];
        const float yi = ysum[1][tid];
        const float nr =  s * yi;                // Re[(-i t/k) y]
        const float ni = -s * yr;                // Im[(-i t/k) y]
        const int row  = rowBase + tid;
        wout_r[row] = nr;
        wout_i[row] = ni;
        psi_r[row] += nr;
        psi_i[row] += ni;
    }
}

// ---------------------------------------------------------------------------
// Born probabilities
// ---------------------------------------------------------------------------
__global__ void __launch_bounds__(256) qnn_probs(const float* __restrict__ psi_r,
                                                 const float* __restrict__ psi_i,
                                                 float* __restrict__ out) {
    const int i = blockIdx.x * 256 + threadIdx.x;
    const float r = psi_r[i], im = psi_i[i];
    out[i] = r * r + im * im;
}

// ---------------------------------------------------------------------------
extern "C" void kernel_launch(void* const* d_in, const int* in_sizes, int n_in,
                              void* d_out, int out_size, void* d_ws, size_t ws_size,
                              hipStream_t stream) {
    const float* feature = (const float*)d_in[0];   // [1024]
    const float* theta   = (const float*)d_in[1];   // [8,1] flat
    const float* gens    = (const float*)d_in[2];   // [8,1024,1024]
    float* out = (float*)d_out;

    float* ws    = (float*)d_ws;        // 6 * 1024 floats = 24 KB scratch
    float* psi_r = ws + 0 * DIM;
    float* psi_i = ws + 1 * DIM;
    float* wa_r  = ws + 2 * DIM;
    float* wa_i  = ws + 3 * DIM;
    float* wb_r  = ws + 4 * DIM;
    float* wb_i  = ws + 5 * DIM;

    qnn_init<<<1, 256, 0, stream>>>(feature, psi_r, psi_i, wa_r, wa_i);

    for (int g = 0; g < NGATES; ++g) {
        const float* G  = gens + (size_t)g * DIM * DIM;
        const float* th = theta + g;

        // w0 = psi  (into buffer A; term k=1 reads A, writes B, then ping-pong)
        qnn_copy<<<DIM / 256, 256, 0, stream>>>(psi_r, psi_i, wa_r, wa_i);

        for (int k = 1; k <= NTERMS; ++k) {
            const bool a_in = (k & 1);   // k=1: read A, write B
            const float* in_r = a_in ? wa_r : wb_r;
            const float* in_i = a_in ? wa_i : wb_i;
            float* out_r = a_in ? wb_r : wa_r;
            float* out_i = a_in ? wb_i : wa_i;
            qnn_term<<<DIM / 16, 256, 0, stream>>>(
                G, th, in_r, in_i, out_r, out_i, psi_r, psi_i, 1.0f / (float)k);
        }
    }

    qnn_probs<<<DIM / 256, 256, 0, stream>>>(psi_r, psi_i, out);
}